// SinkhornDistance_91130616086601
// MI455X (gfx1250) — compile-verified
//
#include <hip/hip_runtime.h>
#include <math.h>

// Sinkhorn distillation loss on gfx1250 (MI455X), B=32, N=1024.
// Log-domain Sinkhorn with global-shift stabilization; per-row exp-sum
// reductions accumulated with V_WMMA_F32_16X16X4_F32 (B = ones),
// dual accumulators to pipeline the matrix unit. Exponentials use the raw
// hardware exp2 (v_exp_f32) via __builtin_amdgcn_exp2f.

typedef __attribute__((ext_vector_type(2))) float v2f;
typedef __attribute__((ext_vector_type(8))) float v8f;

#define BN   32
#define NN   1024
#define EPSR 0.1f
#define INV_EPS 10.0f
#define INV_T 0.1f
#define THRESH 0.1f
#define MAX_ITER 30
#define GRAD_COEF 100.0f
#define LOG_STAB 1e-8f
#define LOG2E 1.4426950408889634f
// exp(w - C/eps) == exp2(w*log2e - (log2e/eps)*C)
#define NEG_IEL2 (-INV_EPS * LOG2E)

#define EXP2_RAW(x) __builtin_amdgcn_exp2f(x)   // bare v_exp_f32

// ---- deterministic block reductions (blockDim.x == 1024) ----
__device__ __forceinline__ float blk_max(float v, float* red) {
    const int tid = threadIdx.x;
    red[tid] = v; __syncthreads();
    #pragma unroll
    for (int s = 512; s > 0; s >>= 1) {
        if (tid < s) red[tid] = fmaxf(red[tid], red[tid + s]);
        __syncthreads();
    }
    float r = red[0]; __syncthreads();
    return r;
}

__device__ __forceinline__ float blk_sum(float v, float* red) {
    const int tid = threadIdx.x;
    red[tid] = v; __syncthreads();
    #pragma unroll
    for (int s = 512; s > 0; s >>= 1) {
        if (tid < s) red[tid] = red[tid] + red[tid + s];
        __syncthreads();
    }
    float r = red[0]; __syncthreads();
    return r;
}

// ---- row-block LSE inner sums via WMMA ----------------------------------
// For 16 consecutive rows per wave:
//   rowsum[row] = sum_j exp((pot[j] - C(row,j) - shift)/eps)
// A-matrix layout for V_WMMA_F32_16X16X4_F32: lane l holds row (l&15),
// K = 2*(l>>4) + vgpr.  B = ones(4x16)  =>  C[i][*] accumulates row sums.
// wscratch (NN floats) holds w[c] = (pot[c]-shift)*(log2e/eps) so the
// per-element exponent is a single FMA feeding v_exp_f32.
__device__ __forceinline__ void lse_rows(const float* __restrict__ feat_row,
                                         const float* __restrict__ feat_col,
                                         const float* __restrict__ pot_col,
                                         float shift,
                                         float* __restrict__ rowsum,
                                         float* __restrict__ wscratch) {
    const int tid  = threadIdx.x;
    const int wv   = tid >> 5;
    const int lane = tid & 31;
    const int mrow = lane & 15;
    const int kbase = (lane >> 4) << 1;

    // hoist shift, eps scaling, and exp->exp2 conversion out of the hot loop
    wscratch[tid] = (pot_col[tid] - shift) * (INV_EPS * LOG2E);
    __syncthreads();

    v2f bones; bones.x = 1.0f; bones.y = 1.0f;

    for (int rb = wv; rb < NN / 16; rb += 32) {
        const int   row  = (rb << 4) + mrow;
        const float fr   = feat_row[row];
        v8f acc0 = {0.f, 0.f, 0.f, 0.f, 0.f, 0.f, 0.f, 0.f};
        v8f acc1 = {0.f, 0.f, 0.f, 0.f, 0.f, 0.f, 0.f, 0.f};

        // strength-reduced (row - col): integers < 2^24 are exact in f32
        float dA = (float)(row - kbase);

        for (int j0 = 0; j0 < NN; j0 += 8) {
            {
                const int c0 = j0 + kbase;
                float d0 = dA;
                float p0 = fr - feat_col[c0];
                float t0 = fmaf(NEG_IEL2, fmaf(d0, d0, p0 * p0), wscratch[c0]);
                float d1 = d0 - 1.0f;
                float p1 = fr - feat_col[c0 + 1];
                float t1 = fmaf(NEG_IEL2, fmaf(d1, d1, p1 * p1), wscratch[c0 + 1]);
                v2f a; a.x = EXP2_RAW(t0); a.y = EXP2_RAW(t1);
                acc0 = __builtin_amdgcn_wmma_f32_16x16x4_f32(
                    false, a, false, bones, (short)0, acc0, false, false);
            }
            {
                const int c0 = j0 + 4 + kbase;
                float d0 = dA - 4.0f;
                float p0 = fr - feat_col[c0];
                float t0 = fmaf(NEG_IEL2, fmaf(d0, d0, p0 * p0), wscratch[c0]);
                float d1 = d0 - 1.0f;
                float p1 = fr - feat_col[c0 + 1];
                float t1 = fmaf(NEG_IEL2, fmaf(d1, d1, p1 * p1), wscratch[c0 + 1]);
                v2f a; a.x = EXP2_RAW(t0); a.y = EXP2_RAW(t1);
                acc1 = __builtin_amdgcn_wmma_f32_16x16x4_f32(
                    false, a, false, bones, (short)0, acc1, false, false);
            }
            dA -= 8.0f;
        }
        // C layout: vgpr r, lanes 0-15 -> row r ; lanes 16-31 -> row 8+r.
        if ((lane & 15) == 0) {
            const int rbase = (rb << 4) + ((lane >> 4) << 3);
            #pragma unroll
            for (int r = 0; r < 8; ++r) rowsum[rbase + r] = acc0[r] + acc1[r];
        }
    }
    __syncthreads();
}

// ---- kernel 0: softmax(x/T), softmax(y/T), zero u/v, clear done ----------
__global__ void __launch_bounds__(NN)
sk_init_kernel(const float* __restrict__ x1,
               const float* __restrict__ y1,
               float* __restrict__ px, float* __restrict__ py,
               float* __restrict__ u,  float* __restrict__ v,
               int* __restrict__ done) {
    __shared__ float red[NN];
    const int b = blockIdx.x, tid = threadIdx.x;
    const int idx = b * NN + tid;

    float t = x1[idx] * INV_T;
    float m = blk_max(t, red);
    float e = EXP2_RAW((t - m) * LOG2E);
    float s = blk_sum(e, red);
    px[idx] = e / s;

    t = y1[idx] * INV_T;
    m = blk_max(t, red);
    e = EXP2_RAW((t - m) * LOG2E);
    s = blk_sum(e, red);
    py[idx] = e / s;

    u[idx] = 0.0f;
    v[idx] = 0.0f;
    if (b == 0 && tid == 0) *done = 0;
}

// ---- kernel 1: one full Sinkhorn iteration per batch (gated on done) -----
__global__ void __launch_bounds__(NN)
sk_iter_kernel(const float* __restrict__ px,
               const float* __restrict__ py,
               float* __restrict__ gu, float* __restrict__ gv,
               float* __restrict__ errArr,
               const int* __restrict__ done) {
    if (*done) return;
    __shared__ float px_l[NN], py_l[NN], u_l[NN], v_l[NN], rowsum[NN], red[NN];
    const int b = blockIdx.x, tid = threadIdx.x;
    const int idx = b * NN + tid;

    px_l[tid] = px[idx];
    py_l[tid] = py[idx];
    u_l[tid]  = gu[idx];
    v_l[tid]  = gv[idx];
    __syncthreads();

    const float cmu = EPSR * __logf(1.0f / (float)NN + LOG_STAB);

    // ---- u half:  u_i = eps*log_mu - eps*LSE_j((v_j - C_ij)/eps) ----
    float s = blk_max(v_l[tid], red);
    lse_rows(px_l, py_l, v_l, s, rowsum, red);
    float un = cmu - s - EPSR * __logf(rowsum[tid]);
    float du = fabsf(un - u_l[tid]);
    u_l[tid] = un;
    gu[idx]  = un;
    float esum = blk_sum(du, red);         // barrier also publishes u_l
    if (tid == 0) errArr[b] = esum;

    // ---- v half:  v_j = eps*log_nu - eps*LSE_i((u_i - C_ij)/eps) ----
    float s2 = blk_max(u_l[tid], red);
    lse_rows(py_l, px_l, u_l, s2, rowsum, red);
    float vn = cmu - s2 - EPSR * __logf(rowsum[tid]);
    gv[idx] = vn;
}

// ---- kernel 2: batch-scalar early-stop flag (deterministic order) --------
__global__ void sk_done_kernel(const float* __restrict__ errArr,
                               int* __restrict__ done) {
    if (threadIdx.x == 0 && !(*done)) {
        float s = 0.0f;
        for (int b = 0; b < BN; ++b) s += errArr[b];
        if (s * (1.0f / (float)BN) < THRESH) *done = 1;
    }
}

// ---- kernel 3: per-batch transport cost sum(pi * C) ----------------------
__global__ void __launch_bounds__(NN)
sk_cost_kernel(const float* __restrict__ px,
               const float* __restrict__ py,
               const float* __restrict__ gu,
               const float* __restrict__ gv,
               float* __restrict__ costArr) {
    __shared__ float py_l[NN], v_l[NN], red[NN];
    const int b = blockIdx.x, tid = threadIdx.x;
    const int idx = b * NN + tid;
    py_l[tid] = py[idx];
    v_l[tid]  = gv[idx];
    __syncthreads();

    const float ui   = gu[idx];
    const float pxi  = px[idx];
    float d   = (float)tid;     // tid - j, strength-reduced
    float acc = 0.0f;
    for (int j = 0; j < NN; ++j) {
        float p = pxi - py_l[j];
        float C = fmaf(d, d, p * p);
        float pi = EXP2_RAW(fmaf(NEG_IEL2, C, (ui + v_l[j]) * (INV_EPS * LOG2E)));
        acc = fmaf(pi, C, acc);
        d -= 1.0f;
    }
    float s = blk_sum(acc, red);
    if (tid == 0) costArr[b] = s;
}

// ---- kernel 4: final scalar, deterministic order -------------------------
__global__ void sk_final_kernel(const float* __restrict__ costArr,
                                float* __restrict__ out) {
    if (threadIdx.x == 0) {
        float s = 0.0f;
        for (int b = 0; b < BN; ++b) s += costArr[b];
        out[0] = (s * (1.0f / (float)BN)) * GRAD_COEF;
    }
}

extern "C" void kernel_launch(void* const* d_in, const int* in_sizes, int n_in,
                              void* d_out, int out_size, void* d_ws, size_t ws_size,
                              hipStream_t stream) {
    const float* x1 = (const float*)d_in[0];
    const float* y1 = (const float*)d_in[1];
    float* out = (float*)d_out;

    // workspace layout (floats)
    float* ws      = (float*)d_ws;
    float* px      = ws;                     // 32*1024
    float* py      = px + BN * NN;           // 32*1024
    float* u       = py + BN * NN;           // 32*1024
    float* v       = u  + BN * NN;           // 32*1024
    float* errArr  = v  + BN * NN;           // 32
    float* costArr = errArr + BN;            // 32
    int*   done    = (int*)(costArr + BN);   // 1

    sk_init_kernel<<<BN, NN, 0, stream>>>(x1, y1, px, py, u, v, done);
    for (int it = 0; it < MAX_ITER; ++it) {
        sk_iter_kernel<<<BN, NN, 0, stream>>>(px, py, u, v, errArr, done);
        sk_done_kernel<<<1, 32, 0, stream>>>(errArr, done);
    }
    sk_cost_kernel<<<BN, NN, 0, stream>>>(px, py, u, v, costArr);
    sk_final_kernel<<<1, 32, 0, stream>>>(costArr, out);
}